// InverseHaarWavelet2D_13400297963653
// MI455X (gfx1250) — compile-verified
//
#include <hip/hip_runtime.h>

// Inverse 2-D Haar wavelet (conv_transpose2d k=s=2, groups=C) for
//   x: [B=16, 4C=256, H=128, W=128] f32,  filters: [4C,1,2,2] f32
//   y: [B=16, C=64, 2H=256, 2W=256] f32
//
// Memory-bound: 537 MB total traffic -> ~23 us floor @ 23.3 TB/s.
// Strategy: 128-bit non-temporal loads/stores (TH_NT, stream-once data),
// scalar-broadcast filter taps, one thread = 4 input pixel-groups
// = 4 output 2x2 blocks (64 B in / 64 B out per thread, balanced).

typedef float v4f __attribute__((ext_vector_type(4)));

namespace {
constexpr int kB = 16;
constexpr int kC = 64;          // groups
constexpr int kH = 128;
constexpr int kW = 128;
constexpr int kW4 = kW / 4;     // float4 units per input row = 32
constexpr int kThreads = 256;   // 8 waves (wave32)
}

__global__ __launch_bounds__(kThreads) void ihaar2d_kernel(
    const float* __restrict__ x,
    const float* __restrict__ filt,
    float* __restrict__ y)
{
    const int c = blockIdx.y;   // group / output channel (uniform per block)
    const int b = blockIdx.z;   // batch (uniform per block)

    // 16 per-channel taps f[c][k][p][q]; uniform across the block so these
    // lower to scalar loads + broadcast. Layout: filters[(4c+k)*4 + p*2 + q].
    const float* f = filt + (size_t)c * 16;
    float t00[4], t01[4], t10[4], t11[4];
#pragma unroll
    for (int k = 0; k < 4; ++k) {
        t00[k] = f[k * 4 + 0];  // (p=0,q=0)
        t01[k] = f[k * 4 + 1];  // (p=0,q=1)
        t10[k] = f[k * 4 + 2];  // (p=1,q=0)
        t11[k] = f[k * 4 + 3];  // (p=1,q=1)
    }

    const int unit = blockIdx.x * kThreads + threadIdx.x;  // 0 .. kH*kW4-1
    const int h  = unit >> 5;            // kW4 == 32
    const int w4 = unit & 31;
    const int w0 = w4 * 4;

    const size_t plane = (size_t)kH * kW;
    const float* xp = x + (((size_t)b * (4 * kC) + 4 * (size_t)c) * plane)
                        + (size_t)h * kW + (size_t)w0;

    // 4 subband vectors for 4 consecutive w positions (global_load_b128 TH_NT).
    const v4f sLL = __builtin_nontemporal_load((const v4f*)(xp + 0 * plane));
    const v4f sLH = __builtin_nontemporal_load((const v4f*)(xp + 1 * plane));
    const v4f sHL = __builtin_nontemporal_load((const v4f*)(xp + 2 * plane));
    const v4f sHH = __builtin_nontemporal_load((const v4f*)(xp + 3 * plane));

    float r0[8];  // output row 2h   : columns 2w0 .. 2w0+7
    float r1[8];  // output row 2h+1 : columns 2w0 .. 2w0+7
#pragma unroll
    for (int j = 0; j < 4; ++j) {
        const float a  = sLL[j];
        const float bb = sLH[j];
        const float cc = sHL[j];
        const float dd = sHH[j];
        r0[2 * j + 0] = a * t00[0] + bb * t00[1] + cc * t00[2] + dd * t00[3];
        r0[2 * j + 1] = a * t01[0] + bb * t01[1] + cc * t01[2] + dd * t01[3];
        r1[2 * j + 0] = a * t10[0] + bb * t10[1] + cc * t10[2] + dd * t10[3];
        r1[2 * j + 1] = a * t11[0] + bb * t11[1] + cc * t11[2] + dd * t11[3];
    }

    const size_t oW = 2 * (size_t)kW;
    float* yp = y + (((size_t)b * kC + (size_t)c) * (4 * plane))
                  + (size_t)(2 * h) * oW + (size_t)(2 * w0);

    const v4f o00 = {r0[0], r0[1], r0[2], r0[3]};
    const v4f o01 = {r0[4], r0[5], r0[6], r0[7]};
    const v4f o10 = {r1[0], r1[1], r1[2], r1[3]};
    const v4f o11 = {r1[4], r1[5], r1[6], r1[7]};

    __builtin_nontemporal_store(o00, (v4f*)(yp + 0));
    __builtin_nontemporal_store(o01, (v4f*)(yp + 4));
    __builtin_nontemporal_store(o10, (v4f*)(yp + oW + 0));
    __builtin_nontemporal_store(o11, (v4f*)(yp + oW + 4));
}

extern "C" void kernel_launch(void* const* d_in, const int* in_sizes, int n_in,
                              void* d_out, int out_size, void* d_ws, size_t ws_size,
                              hipStream_t stream) {
    (void)in_sizes; (void)n_in; (void)out_size; (void)d_ws; (void)ws_size;
    const float* x    = (const float*)d_in[0];
    const float* filt = (const float*)d_in[1];
    float* y          = (float*)d_out;

    // Blocks per (b,c) plane: H * (W/4) / 256 = 128*32/256 = 16.
    dim3 grid((kH * kW4) / kThreads, kC, kB);
    ihaar2d_kernel<<<grid, dim3(kThreads), 0, stream>>>(x, filt, y);
}